// eventEncoder_39711267619029
// MI455X (gfx1250) — compile-verified
//
#include <hip/hip_runtime.h>

// SNN LIF 2-layer conv net, fully fused over T=16 timesteps.
// Tile-persistent blocks: out tile 32x32. Spike maps s1 (36x36) and s2
// (4ch x 34x34) live in LDS (they are spatially shared by the 3x3 convs);
// membrane potentials v1/v2 live in registers (pointwise in time, fixed
// thread->pixel mapping). x tiles streamed via the CDNA5 Tensor Data Mover
// (tensor_load_to_lds), double-buffered, overlapped with compute.

#define T_STEPS 16
#define BATCH   32
#define HH      256
#define WW      256
#define TILE    32
#define EXT1    34   // TILE + 2  (v2/s2 region)
#define EXT2    36   // TILE + 4  (v1/s1/x region)
#define NTHREADS 256

typedef unsigned int u32x4 __attribute__((ext_vector_type(4)));
typedef int          i32x8 __attribute__((ext_vector_type(8)));
typedef int          i32x4 __attribute__((ext_vector_type(4)));

// Build a TDM descriptor (ISA cdna5 §8) for a 2D 36x36 fp32 tile with row
// stride WW, and issue TENSOR_LOAD_TO_LDS. tdim0/tdim1 are the remaining
// tensor extents from the tile origin; origins are clamped so the tile is
// always fully in-bounds.
__device__ __forceinline__ void tdm_load_tile(const float* gptr,
                                              unsigned lds_off,
                                              int tdim0, int tdim1) {
  unsigned long long ga = (unsigned long long)(size_t)gptr;
  u32x4 g0;
  g0[0] = 1u;                                   // count=1, user mode, no gather
  g0[1] = lds_off;                              // lds_addr (bytes)
  g0[2] = (unsigned)(ga & 0xffffffffu);         // global_addr[31:0]
  g0[3] = (unsigned)((ga >> 32) & 0x01ffffffu)  // global_addr[56:32]
        | (2u << 30);                           // type=2 ("image")
  i32x8 g1;
  g1[0] = (int)(2u << 16);                      // data_size=2 -> 4 bytes
  g1[1] = (int)(((unsigned)tdim0 & 0xffffu) << 16);           // tensor_dim0[15:0]
  g1[2] = (int)((((unsigned)tdim0 >> 16) & 0xffffu)
        | (((unsigned)tdim1 & 0xffffu) << 16));               // dim0 hi | dim1 lo
  g1[3] = (int)((((unsigned)tdim1 >> 16) & 0xffffu)
        | ((unsigned)EXT2 << 16));                            // dim1 hi | tile_dim0=36
  g1[4] = (int)EXT2;                            // tile_dim1=36, tile_dim2=0 (2D)
  g1[5] = WW;                                   // tensor_dim0_stride[31:0] = 256
  g1[6] = 0;                                    // stride0 hi | stride1 lo
  g1[7] = 0;                                    // stride1 hi
  i32x4 z4 = {0, 0, 0, 0};                      // groups 2/3 unused (2D tensor)
  i32x8 z8 = {0, 0, 0, 0, 0, 0, 0, 0};
  __builtin_amdgcn_tensor_load_to_lds(g0, g1, z4, z4, z8, 0);
}

__global__ void __launch_bounds__(NTHREADS)
snn_fused_kernel(const float* __restrict__ x,
                 const float* __restrict__ w1,
                 const float* __restrict__ w2,
                 float* __restrict__ out)
{
  __shared__ float xbuf[2][EXT2 * EXT2];   // double-buffered x tile (TDM dest)
  __shared__ float s1s[EXT2 * EXT2];       // layer-1 spikes
  __shared__ float s2s[4][EXT1 * EXT1];    // layer-2 spikes

  const int tid = threadIdx.x;
  const int b   = blockIdx.z;
  const int ty0 = blockIdx.y * TILE;
  const int tx0 = blockIdx.x * TILE;

  // Clamped x-tile origin (keeps the 36x36 TDM tile fully inside the image;
  // out-of-image halo pixels are masked to spike=0, i.e. SAME zero padding).
  int ox = tx0 - 2; ox = ox < 0 ? 0 : ox; ox = ox > (WW - EXT2) ? (WW - EXT2) : ox;
  int oy = ty0 - 2; oy = oy < 0 ? 0 : oy; oy = oy > (HH - EXT2) ? (HH - EXT2) : oy;

  // Weights: uniform address + constant offsets -> scalar loads.
  float W1[36], W2[36];
  #pragma unroll
  for (int i = 0; i < 36; ++i) { W1[i] = w1[i]; W2[i] = w2[i]; }

  // Prefetch x tile for t=0 (wave 0 issues the DMA; EXEC-independent op).
  if (tid < 32) {
    tdm_load_tile(x + ((size_t)b) * HH * WW + (size_t)oy * WW + ox,
                  (unsigned)(size_t)&xbuf[0][0], WW - ox, HH - oy);
  }

  // ---------- t-invariant precompute: Phase A (v1/s1, 36x36 strided) -------
  int  aidx[6];   // x-tile LDS index (clamped, safe to load speculatively)
  int  asi[6];    // s1 store index
  bool aval[6];   // i < 36*36
  bool ain[6];    // pixel inside image
  float v1r[6] = {0.f, 0.f, 0.f, 0.f, 0.f, 0.f};
  #pragma unroll
  for (int ii = 0; ii < 6; ++ii) {
    int i = tid + ii * NTHREADS;
    aval[ii] = (i < EXT2 * EXT2);
    int ic = aval[ii] ? i : (EXT2 * EXT2 - 1);
    int ly = ic / EXT2, lx = ic - ly * EXT2;
    int gy = ty0 - 2 + ly, gx = tx0 - 2 + lx;
    ain[ii] = (gy >= 0) && (gy < HH) && (gx >= 0) && (gx < WW);
    int ry = gy - oy; ry = ry < 0 ? 0 : (ry > EXT2 - 1 ? EXT2 - 1 : ry);
    int rx = gx - ox; rx = rx < 0 ? 0 : (rx > EXT2 - 1 ? EXT2 - 1 : rx);
    aidx[ii] = ry * EXT2 + rx;
    asi[ii]  = ic;
  }

  // ---------- t-invariant precompute: Phase B (v2/s2, 34x34) ---------------
  // Main: 16x16 thread grid, 2x2 pixels each, covering rows/cols [0,31].
  const int br = (tid >> 4) << 1;          // 0,2,..,30
  const int bc = (tid & 15) << 1;
  bool bin[2][2];
  #pragma unroll
  for (int dy = 0; dy < 2; ++dy)
    #pragma unroll
    for (int dx = 0; dx < 2; ++dx) {
      int gy = ty0 - 1 + br + dy, gx = tx0 - 1 + bc + dx;
      bin[dy][dx] = (gy >= 0) && (gy < HH) && (gx >= 0) && (gx < WW);
    }
  float v2m[2][2][4] = {};
  // Border: rows 32,33 (68 px) + cols 32,33 of rows 0..31 (64 px) = 132 px,
  // one per thread for tid < 132.
  const bool hasb = (tid < 132);
  int bly = 0, blx = 0;
  if (tid < 68)       { bly = 32 + tid / 34; blx = tid % 34; }
  else if (tid < 132) { int k = tid - 68; bly = k >> 1; blx = 32 + (k & 1); }
  bool binb;
  {
    int gy = ty0 - 1 + bly, gx = tx0 - 1 + blx;
    binb = hasb && (gy >= 0) && (gy < HH) && (gx >= 0) && (gx < WW);
  }
  float v2b[4] = {0.f, 0.f, 0.f, 0.f};

  // Phase C: same 2x2 block as Phase B main (out pixel (br+dy, bc+dx)).
  float acc00 = 0.f, acc01 = 0.f, acc10 = 0.f, acc11 = 0.f;

  if (tid < 32) __builtin_amdgcn_s_wait_tensorcnt(0);
  __syncthreads();

  int cur = 0;
  for (int t = 0; t < T_STEPS; ++t) {
    const int nxt = cur ^ 1;

    // Kick off DMA of next timestep's tile; overlaps with compute below.
    if ((t + 1 < T_STEPS) && (tid < 32)) {
      tdm_load_tile(x + ((size_t)(t + 1) * BATCH + b) * HH * WW
                      + (size_t)oy * WW + ox,
                    (unsigned)(size_t)&xbuf[nxt][0], WW - ox, HH - oy);
    }

    // ---- Phase A: v1 integrate + spike (v1 in regs, s1 -> LDS) ------------
    {
      const float* xb = &xbuf[cur][0];
      #pragma unroll
      for (int ii = 0; ii < 6; ++ii) {
        if (aval[ii]) {
          float xv  = ain[ii] ? xb[aidx[ii]] : 0.0f;
          float v1  = v1r[ii];
          float v1n = v1 + (xv - v1) * 0.5f;          // tau = 2
          bool  sp  = ain[ii] && (v1n >= 1.0f);       // atan_spike fwd
          s1s[asi[ii]] = sp ? 1.0f : 0.0f;
          v1r[ii]      = sp ? 0.0f : v1n;             // hard reset
        }
      }
    }
    __syncthreads();

    // ---- Phase B: conv1 (1->4ch, 3x3) + v2 integrate + spike --------------
    // Main 2x2 block: 4x4 s1 window shared by the 4 pixels.
    {
      float a[4][4];
      #pragma unroll
      for (int rr = 0; rr < 4; ++rr)
        #pragma unroll
        for (int cc = 0; cc < 4; ++cc)
          a[rr][cc] = s1s[(br + rr) * EXT2 + (bc + cc)];
      #pragma unroll
      for (int dy = 0; dy < 2; ++dy)
        #pragma unroll
        for (int dx = 0; dx < 2; ++dx) {
          float c0 = 0.f, c1 = 0.f, c2 = 0.f, c3 = 0.f;
          #pragma unroll
          for (int ky = 0; ky < 3; ++ky)
            #pragma unroll
            for (int kx = 0; kx < 3; ++kx) {
              float sv = a[dy + ky][dx + kx];
              int   wi = ky * 3 + kx;
              c0 += W1[0 * 9 + wi] * sv;
              c1 += W1[1 * 9 + wi] * sv;
              c2 += W1[2 * 9 + wi] * sv;
              c3 += W1[3 * 9 + wi] * sv;
            }
          const int si = (br + dy) * EXT1 + (bc + dx);
          float cc4[4] = {c0, c1, c2, c3};
          #pragma unroll
          for (int ch = 0; ch < 4; ++ch) {
            float v = v2m[dy][dx][ch];
            v = v + (cc4[ch] - v) * 0.5f;
            bool sp = bin[dy][dx] && (v >= 1.0f);
            s2s[ch][si]     = sp ? 1.0f : 0.0f;
            v2m[dy][dx][ch] = sp ? 0.0f : v;
          }
        }
    }
    // Border pixel (one per thread for tid < 132).
    if (hasb) {
      float c0 = 0.f, c1 = 0.f, c2 = 0.f, c3 = 0.f;
      #pragma unroll
      for (int ky = 0; ky < 3; ++ky)
        #pragma unroll
        for (int kx = 0; kx < 3; ++kx) {
          float sv = s1s[(bly + ky) * EXT2 + (blx + kx)];
          int   wi = ky * 3 + kx;
          c0 += W1[0 * 9 + wi] * sv;
          c1 += W1[1 * 9 + wi] * sv;
          c2 += W1[2 * 9 + wi] * sv;
          c3 += W1[3 * 9 + wi] * sv;
        }
      const int si = bly * EXT1 + blx;
      float cc4[4] = {c0, c1, c2, c3};
      #pragma unroll
      for (int ch = 0; ch < 4; ++ch) {
        float v = v2b[ch];
        v = v + (cc4[ch] - v) * 0.5f;
        bool sp = binb && (v >= 1.0f);
        s2s[ch][si] = sp ? 1.0f : 0.0f;
        v2b[ch]     = sp ? 0.0f : v;
      }
    }
    __syncthreads();

    // ---- Phase C: conv2 (4->1ch, 3x3), 2x2 register-blocked ---------------
    {
      float o00 = 0.f, o01 = 0.f, o10 = 0.f, o11 = 0.f;
      #pragma unroll
      for (int ch = 0; ch < 4; ++ch) {
        float r[4][4];
        #pragma unroll
        for (int rr = 0; rr < 4; ++rr)
          #pragma unroll
          for (int cc = 0; cc < 4; ++cc)
            r[rr][cc] = s2s[ch][(br + rr) * EXT1 + (bc + cc)];
        #pragma unroll
        for (int ky = 0; ky < 3; ++ky)
          #pragma unroll
          for (int kx = 0; kx < 3; ++kx) {
            float w = W2[ch * 9 + ky * 3 + kx];
            o00 += w * r[ky    ][kx    ];
            o01 += w * r[ky    ][kx + 1];
            o10 += w * r[ky + 1][kx    ];
            o11 += w * r[ky + 1][kx + 1];
          }
      }
      acc00 += o00; acc01 += o01; acc10 += o10; acc11 += o11;
    }

    // Make next x tile visible to everyone before swapping buffers.
    if (tid < 32) __builtin_amdgcn_s_wait_tensorcnt(0);
    __syncthreads();
    cur = nxt;
  }

  // Mean over T, write the 2x2 out block (out pixel = (ty0+br, tx0+bc)+...).
  {
    const float inv = 1.0f / (float)T_STEPS;
    size_t base = ((size_t)b * HH + (ty0 + br)) * WW + (tx0 + bc);
    out[base]          = acc00 * inv;
    out[base + 1]      = acc01 * inv;
    out[base + WW]     = acc10 * inv;
    out[base + WW + 1] = acc11 * inv;
  }
}

extern "C" void kernel_launch(void* const* d_in, const int* in_sizes, int n_in,
                              void* d_out, int out_size, void* d_ws, size_t ws_size,
                              hipStream_t stream) {
  const float* x  = (const float*)d_in[0];   // (16, 32, 256, 256) f32
  const float* w1 = (const float*)d_in[1];   // (4, 1, 3, 3) f32
  const float* w2 = (const float*)d_in[2];   // (1, 4, 3, 3) f32
  float* out = (float*)d_out;                // (32, 1, 256, 256) f32

  dim3 grid(WW / TILE, HH / TILE, BATCH);    // 8 x 8 x 32 = 2048 blocks
  snn_fused_kernel<<<grid, NTHREADS, 0, stream>>>(x, w1, w2, out);
}